// SobelLoss_8143257993975
// MI455X (gfx1250) — compile-verified
//
#include <hip/hip_runtime.h>

// Problem geometry (fixed by the reference): [B=2, 1, D=160, H=192, W=160] f32
#define W_ 160
#define H_ 192
#define D_ 160
#define B_ 2
#define WH_ (W_ * H_)            // 30720
#define WHD_ (W_ * H_ * D_)      // 4915200
#define NVOX ((double)B_ * WHD_) // 9830400

// Tile: 32 x 8 x 16 outputs per 256-thread block (all dims divide exactly)
#define TX 32
#define TY 8
#define TZ 16
#define IX (TX + 2)   // 34
#define IY (TY + 2)   // 10
#define IZ (TZ + 2)   // 18
#define LSX 36                    // padded x-stride in LDS
#define LSLICE (LSX * IY)         // 360
#define LDS_ELEMS (LSLICE * IZ)   // 6480 floats per buffer
#define NTX (W_ / TX)  // 5
#define NTY (H_ / TY)  // 24
#define NTZ (D_ / TZ)  // 10
#define NBLOCKS (NTX * NTY * NTZ * B_)  // 2400

// ---- CDNA5 async global->LDS path (ASYNCcnt), probe-verified present ----
#if defined(__has_builtin)
#if __has_builtin(__builtin_amdgcn_global_load_async_to_lds_b32)
#define HAVE_ASYNC_LDS 1
#endif
#if __has_builtin(__builtin_amdgcn_s_wait_asynccnt)
#define HAVE_WAIT_ASYNC 1
#endif
#endif

typedef __attribute__((address_space(1))) int as1_int;
typedef __attribute__((address_space(3))) int as3_int;

__device__ __forceinline__ void async_ld_f32(const float* g, float* l) {
#ifdef HAVE_ASYNC_LDS
  __builtin_amdgcn_global_load_async_to_lds_b32(
      (as1_int*)g, (as3_int*)l, 0 /*imm offset*/, 0 /*cpol*/);
#else
  *l = __builtin_nontemporal_load(g);
#endif
}

__device__ __forceinline__ void wait_async_lds() {
#ifdef HAVE_ASYNC_LDS
#ifdef HAVE_WAIT_ASYNC
  __builtin_amdgcn_s_wait_asynccnt(0);
#else
  asm volatile("s_wait_asynccnt 0" ::: "memory");
#endif
#endif
}

__global__ void sobel_init_kernel(double* acc) {
  if (threadIdx.x == 0 && blockIdx.x == 0) *acc = 0.0;
}

__global__ __launch_bounds__(256) void sobel_loss_kernel(
    const float* __restrict__ moved, const float* __restrict__ label,
    double* __restrict__ acc) {
  __shared__ float sM[LDS_ELEMS];
  __shared__ float sL[LDS_ELEMS];
  __shared__ float sWave[8];

  const int tid = threadIdx.x;
  int id = blockIdx.x;
  const int bx = id % NTX; id /= NTX;
  const int by = id % NTY; id /= NTY;
  const int bz = id % NTZ; id /= NTZ;
  const int bb = id;

  const bool border = (bx == 0) | (bx == NTX - 1) | (by == 0) |
                      (by == NTY - 1) | (bz == 0) | (bz == NTZ - 1);

  // Zero LDS tiles only when the tile touches a volume border: gives free
  // zero-padding there. Interior tiles overwrite every logical cell.
  if (border) {
    for (int i = tid; i < LDS_ELEMS; i += 256) { sM[i] = 0.0f; sL[i] = 0.0f; }
    __syncthreads();  // all zero-stores visible before async writes land
  }

  const int x0 = bx * TX - 1;
  const int y0 = by * TY - 1;
  const int z0 = bz * TZ - 1;
  const int gbase = bb * WHD_;

  // Async-copy the 34x10x18 halo tiles of both tensors straight into LDS
  // (no VGPR staging; tracked with ASYNCcnt).
  for (int i = tid; i < IX * IY * IZ; i += 256) {
    const int lx = i % IX;
    const int r  = i / IX;
    const int ly = r % IY;
    const int lz = r / IY;
    const int gx = x0 + lx, gy = y0 + ly, gz = z0 + lz;
    if ((unsigned)gx < (unsigned)W_ && (unsigned)gy < (unsigned)H_ &&
        (unsigned)gz < (unsigned)D_) {
      const int gi = gbase + gz * WH_ + gy * W_ + gx;
      const int li = lz * LSLICE + ly * LSX + lx;
      async_ld_f32(&moved[gi], &sM[li]);
      async_ld_f32(&label[gi], &sL[li]);
    }
  }
  wait_async_lds();
  __syncthreads();

  // d = moved - label, in place (conv is linear; do it once in LDS).
  for (int i = tid; i < LDS_ELEMS; i += 256) sM[i] -= sL[i];
  __syncthreads();

  // Separable Sobel: per input layer compute U = (s_y (x) d_x),
  // V = (d_y (x) s_x), Wq = (s_y (x) s_x); combine along z with a 3-deep
  // register ring (fully unrolled -> ring shifts become register renames).
  const int tx = tid % TX;  // output x within tile
  const int ty = tid / TX;  // output y within tile
  const float* base = &sM[ty * LSX + tx];

  float u0 = 0.f, u1 = 0.f, u2 = 0.f;
  float v0 = 0.f, v1 = 0.f, v2 = 0.f;
  float w0 = 0.f, w1 = 0.f, w2 = 0.f;
  float accf = 0.f;

#pragma unroll
  for (int lz = 0; lz < IZ; ++lz) {
    const float* p = base + lz * LSLICE;
    float a = p[0],         b = p[1],           c = p[2];
    const float rs0 = (a + c) + 2.f * b, rd0 = c - a;
    a = p[LSX];     b = p[LSX + 1];     c = p[LSX + 2];
    const float rs1 = (a + c) + 2.f * b, rd1 = c - a;
    a = p[2 * LSX]; b = p[2 * LSX + 1]; c = p[2 * LSX + 2];
    const float rs2 = (a + c) + 2.f * b, rd2 = c - a;

    const float U  = (rd0 + rd2) + 2.f * rd1;  // s_y * d_x
    const float V  = rs2 - rs0;                // d_y * s_x
    const float Wq = (rs0 + rs2) + 2.f * rs1;  // s_y * s_x

    u0 = u1; u1 = u2; u2 = U;
    v0 = v1; v1 = v2; v2 = V;
    w0 = w1; w1 = w2; w2 = Wq;

    if (lz >= 2) {
      const float gx = (u0 + u2) + 2.f * u1;   // s_z
      const float gy = (v0 + v2) + 2.f * v1;   // s_z
      const float gz = w0 - w2;                // d_z
      accf += __builtin_fabsf(gx);
      accf += __builtin_fabsf(gy);
      accf += __builtin_fabsf(gz);
    }
  }

  // Wave32 shuffle reduction, then 8 wave-partials via LDS, one f64 atomic.
  for (int off = 16; off > 0; off >>= 1)
    accf += __shfl_down(accf, off, 32);
  if ((tid & 31) == 0) sWave[tid >> 5] = accf;
  __syncthreads();
  if (tid == 0) {
    float s = sWave[0];
    for (int w = 1; w < 8; ++w) s += sWave[w];
    atomicAdd(acc, (double)s);
  }
}

__global__ void sobel_final_kernel(const double* __restrict__ acc,
                                   float* __restrict__ out) {
  if (threadIdx.x == 0 && blockIdx.x == 0)
    out[0] = (float)(*acc / (3.0 * NVOX));
}

extern "C" void kernel_launch(void* const* d_in, const int* in_sizes, int n_in,
                              void* d_out, int out_size, void* d_ws, size_t ws_size,
                              hipStream_t stream) {
  const float* moved = (const float*)d_in[0];
  const float* label = (const float*)d_in[1];
  double* acc = (double*)d_ws;
  float* out = (float*)d_out;

  sobel_init_kernel<<<1, 32, 0, stream>>>(acc);
  sobel_loss_kernel<<<NBLOCKS, 256, 0, stream>>>(moved, label, acc);
  sobel_final_kernel<<<1, 32, 0, stream>>>(acc, out);
}